// HGTEncoder_2568390443400
// MI455X (gfx1250) — compile-verified
//
#include <hip/hip_runtime.h>
#include <math.h>

typedef float v2f __attribute__((ext_vector_type(2)));
typedef float v8f __attribute__((ext_vector_type(8)));

#define HID   128
#define HEADS 4
#define DH    32
#define EPER  300000

__device__ __forceinline__ float geluf(float x) {
  return 0.5f * x * (1.0f + erff(x * 0.70710678118654752f));
}
__device__ __forceinline__ float leakyf(float x) {
  return x > 0.0f ? x : 0.01f * x;
}
__device__ __forceinline__ void atomicMaxF(float* addr, float v) {
  unsigned int* u = (unsigned int*)addr;
  unsigned int old = *u;
  while (__uint_as_float(old) < v) {
    unsigned int assumed = old;
    old = atomicCAS(u, assumed, __float_as_uint(v));
    if (old == assumed) break;
  }
}

// ---------------------------------------------------------------------------
// Up-front weight repack into WMMA B-fragment order.
// Full W (128 x NT*16):  R[((kb*NT + t)*32 + lane)*2 + {0,1}]
//                          = W[kb*4 + (lane>>4)*2 + {0,1}][t*16 + (lane&15)]
// Rel (4,32,32):         R[(((h*8+kb)*2+th)*32 + lane)*2 + {0,1}]
//                          = rel[h][kb*4 + (lane>>4)*2 + {0,1}][th*16 + (lane&15)]
// One b64 coalesced load per WMMA B operand afterwards.
// ---------------------------------------------------------------------------
struct RepackJobs {
  const float* src[96];
  unsigned     dstOff[96];
  int          kind[96];   // 0: full NT=8, 1: full NT=4, 2: rel
  int          njobs;
};

__global__ __launch_bounds__(256)
void repack_all_kernel(RepackJobs J, float* __restrict__ base)
{
  const int j = blockIdx.y;
  const int idx = blockIdx.x * 256 + threadIdx.x;
  const int kind = J.kind[j];
  const float* W = J.src[j];
  float* R = base + J.dstOff[j];
  if (kind == 2) {
    if (idx >= 2048) return;
    int lane = idx & 31, th = (idx >> 5) & 1, kb = (idx >> 6) & 7, h = idx >> 9;
    int d = kb * 4 + (lane >> 4) * 2, f = th * 16 + (lane & 15);
    const float* Rh = W + h * (DH * DH);
    R[idx * 2]     = Rh[d * DH + f];
    R[idx * 2 + 1] = Rh[(d + 1) * DH + f];
  } else {
    const int NT = (kind == 0) ? 8 : 4;
    const int N  = NT * 16;
    if (idx >= 32 * NT * 32) return;
    int lane = idx & 31, t = (idx >> 5) % NT, kb = idx / (32 * NT);
    int kA = kb * 4 + (lane >> 4) * 2, f = t * 16 + (lane & 15);
    R[idx * 2]     = W[(size_t)kA * N + f];
    R[idx * 2 + 1] = W[(size_t)(kA + 1) * N + f];
  }
}

// ---------------------------------------------------------------------------
// Generic WMMA fp32 GEMM: C[M x (NT*16)] = preop(A[M x 128]) @ W + b
// W pre-repacked: one coalesced global_load_b64 per B fragment.
// PRE: 0=none, 1=exact gelu on A.  EPI: 0=store C, 1=HGT skip-combine into X.
// ---------------------------------------------------------------------------
template<int NT, int PRE, int EPI>
__global__ __launch_bounds__(128)
void wmma_gemm_kernel(const float* __restrict__ A,
                      const v2f* __restrict__ Wr,
                      const float* __restrict__ bias,
                      float* __restrict__ C,
                      float* __restrict__ Xres,
                      const float* __restrict__ skip,
                      int M)
{
  const int lane   = threadIdx.x & 31;
  const int wave   = threadIdx.x >> 5;
  const int rowblk = blockIdx.x * 4 + wave;
  if (rowblk * 16 >= M) return;            // wave-uniform: EXEC all-ones for WMMA
  const int row0 = rowblk * 16;
  const int m    = lane & 15;
  const int koff = (lane >> 4) * 2;        // lanes 0-15: k+0/+1; 16-31: k+2/+3
  const int fcol = lane & 15;
  const int N    = NT * 16;

  v8f acc[NT] = {};
  const float* arow = A + (size_t)(row0 + m) * HID;

  for (int kb = 0; kb < 32; ++kb) {
    const int kA = kb * 4 + koff;
    float ax = arow[kA];
    float ay = arow[kA + 1];
    if (PRE == 1) { ax = geluf(ax); ay = geluf(ay); }
    v2f a; a.x = ax; a.y = ay;
#pragma unroll
    for (int t = 0; t < NT; ++t) {
      v2f b = Wr[(kb * NT + t) * 32 + lane];
      acc[t] = __builtin_amdgcn_wmma_f32_16x16x4_f32(false, a, false, b,
                                                     (short)0, acc[t], false, false);
    }
  }

  float beta = 0.0f;
  if (EPI == 1) beta = 1.0f / (1.0f + __expf(-skip[0]));

#pragma unroll
  for (int t = 0; t < NT; ++t) {
    const int col = t * 16 + fcol;
    const float bv = bias[col];
#pragma unroll
    for (int r = 0; r < 8; ++r) {
      const int mrow = row0 + r + (lane >> 4) * 8;   // C layout per ISA §7.12.2
      float val = acc[t][r] + bv;
      if (EPI == 0) {
        C[(size_t)mrow * N + col] = val;
      } else {
        float xr = Xres[(size_t)mrow * HID + col];
        float nv = beta * val + (1.0f - beta) * xr;  // sigmoid-skip blend
        Xres[(size_t)mrow * HID + col] = leakyf(nv) + xr;  // next-layer x
      }
    }
  }
}

// ---------------------------------------------------------------------------
// Block-diagonal WMMA GEMM: C[M x 128] = A[M x 128] @ blockdiag(rel[h] 32x32)
// (hoisted per-edge-type a_rel / m_rel transform of K / V). rel pre-repacked.
// ---------------------------------------------------------------------------
__global__ __launch_bounds__(128)
void wmma_bdiag_kernel(const float* __restrict__ A,
                       const v2f* __restrict__ Rr,
                       float* __restrict__ C,
                       int M)
{
  const int lane   = threadIdx.x & 31;
  const int wave   = threadIdx.x >> 5;
  const int rowblk = blockIdx.x * 4 + wave;
  if (rowblk * 16 >= M) return;
  const int row0 = rowblk * 16;
  const int m    = lane & 15;
  const int koff = (lane >> 4) * 2;
  const int fcol = lane & 15;

  v8f acc[8] = {};
  const float* arow = A + (size_t)(row0 + m) * HID;

#pragma unroll
  for (int h = 0; h < HEADS; ++h) {
    for (int kb = 0; kb < 8; ++kb) {
      const int d = kb * 4 + koff;
      v2f a; a.x = arow[h * DH + d]; a.y = arow[h * DH + d + 1];
#pragma unroll
      for (int th = 0; th < 2; ++th) {
        const int t = h * 2 + th;
        v2f b = Rr[(((h * 8 + kb) * 2 + th) * 32) + lane];
        acc[t] = __builtin_amdgcn_wmma_f32_16x16x4_f32(false, a, false, b,
                                                       (short)0, acc[t], false, false);
      }
    }
  }

#pragma unroll
  for (int t = 0; t < 8; ++t) {
    const int col = t * 16 + fcol;
#pragma unroll
    for (int r = 0; r < 8; ++r) {
      const int mrow = row0 + r + (lane >> 4) * 8;
      C[(size_t)mrow * HID + col] = acc[t][r];
    }
  }
}

// ---------------------------------------------------------------------------
// Input projection: X = leaky_relu(x_in @ W(F x 128) + b). F <= 9, pure VALU.
// ---------------------------------------------------------------------------
__global__ void input_proj_kernel(const float* __restrict__ xin,
                                  const float* __restrict__ w,
                                  const float* __restrict__ b,
                                  float* __restrict__ X, int M, int F)
{
  int idx = blockIdx.x * blockDim.x + threadIdx.x;
  if (idx >= M * HID) return;
  int n = idx >> 7, j = idx & 127;
  float s = b[j];
  for (int i = 0; i < F; ++i) s += xin[(size_t)n * F + i] * w[i * HID + j];
  X[idx] = leakyf(s);
}

__global__ void init_kernel(float* __restrict__ AGG, float* __restrict__ AMAX,
                            float* __restrict__ ASUM, int nAgg, int nStat)
{
  int i = blockIdx.x * blockDim.x + threadIdx.x;
  if (i < nAgg)  AGG[i] = 0.0f;
  if (i < nStat) { AMAX[i] = -INFINITY; ASUM[i] = 0.0f; }
}

// Pass A: logits a[e,h] = (q[dst,h] . ktilde[src,h]) * p_rel[h] / sqrt(DH); seg-max
__global__ __launch_bounds__(256)
void edge_logits_kernel(const int* __restrict__ ei,
                        const float* __restrict__ KT,
                        const float* __restrict__ Q, int dstOff,
                        const float* __restrict__ prel,
                        float* __restrict__ alpha,
                        float* __restrict__ AMAX)
{
  int idx = blockIdx.x * blockDim.x + threadIdx.x;
  if (idx >= EPER * HEADS) return;
  int e = idx >> 2, h = idx & 3;
  int s = ei[e], d = ei[EPER + e];
  const float4* kt4 = (const float4*)(KT + (size_t)s * HID + h * DH);
  const float4* q4  = (const float4*)(Q + (size_t)(dstOff + d) * HID + h * DH);
  float acc = 0.0f;
#pragma unroll
  for (int i = 0; i < DH / 4; ++i) {
    float4 a = kt4[i], b = q4[i];
    acc += a.x * b.x + a.y * b.y + a.z * b.z + a.w * b.w;
  }
  acc *= prel[h] * 0.17677669529663687f;   // 1/sqrt(32)
  alpha[idx] = acc;
  atomicMaxF(&AMAX[(size_t)(dstOff + d) * HEADS + h], acc);
}

// Pass B: alpha <- exp(alpha - amax[d]); seg-sum
__global__ __launch_bounds__(256)
void edge_softmax_kernel(const int* __restrict__ ei, int dstOff,
                         float* __restrict__ alpha,
                         const float* __restrict__ AMAX,
                         float* __restrict__ ASUM)
{
  int idx = blockIdx.x * blockDim.x + threadIdx.x;
  if (idx >= EPER * HEADS) return;
  int e = idx >> 2, h = idx & 3;
  int g = (dstOff + ei[EPER + e]) * HEADS + h;
  float w = __expf(alpha[idx] - AMAX[g]);
  alpha[idx] = w;
  atomicAdd(&ASUM[g], w);
}

// Pass C: agg[dst] += (alpha / asum[dst]) * vtilde[src]
__global__ __launch_bounds__(256)
void edge_agg_kernel(const int* __restrict__ ei, int dstOff,
                     const float* __restrict__ alpha,
                     const float* __restrict__ ASUM,
                     const float* __restrict__ VT,
                     float* __restrict__ AGG)
{
  int idx = blockIdx.x * blockDim.x + threadIdx.x;
  if (idx >= EPER * HEADS) return;
  int e = idx >> 2, h = idx & 3;
  int s = ei[e], d = ei[EPER + e];
  int g = (dstOff + d) * HEADS + h;
  float w = alpha[idx] / (ASUM[g] + 1e-16f);
  const float4* vt4 = (const float4*)(VT + (size_t)s * HID + h * DH);
  float* ag = AGG + (size_t)(dstOff + d) * HID + h * DH;
#pragma unroll
  for (int i = 0; i < DH / 4; ++i) {
    float4 v = vt4[i];
    atomicAdd(&ag[4 * i + 0], w * v.x);
    atomicAdd(&ag[4 * i + 1], w * v.y);
    atomicAdd(&ag[4 * i + 2], w * v.z);
    atomicAdd(&ag[4 * i + 3], w * v.w);
  }
}

// ---------------------------------------------------------------------------
extern "C" void kernel_launch(void* const* d_in, const int* in_sizes, int n_in,
                              void* d_out, int out_size, void* d_ws, size_t ws_size,
                              hipStream_t stream)
{
  (void)in_sizes; (void)n_in; (void)out_size; (void)ws_size;
  static const int CNT[4]   = {100000, 20000, 10000, 20000};
  static const int FEATN[4] = {6, 4, 4, 9};
  static const int OFF[4]   = {0, 100000, 120000, 130000};
  const int NTOT = 150000;
  // edge types in input order (d_in[4..9]); rel params sorted alphabetically
  static const int ESRC[6]  = {0, 1, 0, 2, 0, 3};
  static const int EDST[6]  = {1, 0, 2, 0, 3, 0};
  static const int RSLOT[6] = {1, 3, 2, 4, 0, 5};

  const float* xin[4];
  for (int t = 0; t < 4; ++t) xin[t] = (const float*)d_in[t];
  const int* ei[6];
  for (int t = 0; t < 6; ++t) ei[t] = (const int*)d_in[4 + t];
  auto P = [&](int i) { return (const float*)d_in[i]; };
  // params flattened (sorted pytree): convs[3] then in_b, in_w, out_b, out_w
  // per conv (54): a_rel[6],k_b[4],k_w[4],m_rel[6],o_b[4],o_w[4],p_rel[6],
  //                q_b[4],q_w[4],skip[4],v_b[4],v_w[4]

  float* ws  = (float*)d_ws;
  float* X   = ws;                                 // 150000*128
  float* Kb  = X   + (size_t)NTOT * HID;
  float* Qb  = Kb  + (size_t)NTOT * HID;
  float* Vb  = Qb  + (size_t)NTOT * HID;
  float* AGG = Vb  + (size_t)NTOT * HID;
  float* TIL = AGG + (size_t)NTOT * HID;           // 100000*128 (max src count)
  float* ALP = TIL + (size_t)100000 * HID;         // 6*300000*4
  float* AMX = ALP + (size_t)6 * EPER * HEADS;     // 150000*4
  float* ASM = AMX + (size_t)NTOT * HEADS;         // 150000*4
  float* RPK = ASM + (size_t)NTOT * HEADS;         // repacked weights (~3.9MB)

  // ---- build repack job table (all weights, all layers, once) ----
  RepackJobs J; J.njobs = 0;
  size_t roff = 0;
  auto addJob = [&](const float* src, int kind) -> const v2f* {
    J.src[J.njobs] = src; J.dstOff[J.njobs] = (unsigned)roff; J.kind[J.njobs] = kind;
    const v2f* p = (const v2f*)(RPK + roff);
    roff += (kind == 0) ? 16384 : (kind == 1) ? 8192 : 4096;
    J.njobs++;
    return p;
  };
  const v2f *kwR[3][4], *qwR[3][4], *vwR[3][4], *owR[3][4], *outR[4];
  const v2f *arR[3][6], *mrR[3][6];
  for (int L = 0; L < 3; ++L) {
    const int cb = 10 + L * 54;
    for (int t = 0; t < 4; ++t) {
      kwR[L][t] = addJob(P(cb + 10 + t), 0);
      qwR[L][t] = addJob(P(cb + 38 + t), 0);
      vwR[L][t] = addJob(P(cb + 50 + t), 0);
      owR[L][t] = addJob(P(cb + 24 + t), 0);
    }
    for (int s = 0; s < 6; ++s) {
      arR[L][s] = addJob(P(cb + s), 2);         // a_rel sorted slot s
      mrR[L][s] = addJob(P(cb + 14 + s), 2);    // m_rel sorted slot s
    }
  }
  for (int t = 0; t < 4; ++t) outR[t] = addJob(P(184 + t), 1);

  repack_all_kernel<<<dim3(32, J.njobs), 256, 0, stream>>>(J, RPK);

  const int egrid = (EPER * HEADS + 255) / 256;

  // ---- input projections ----
  for (int t = 0; t < 4; ++t) {
    int n = CNT[t] * HID;
    input_proj_kernel<<<(n + 255) / 256, 256, 0, stream>>>(
        xin[t], P(176 + t), P(172 + t), X + (size_t)OFF[t] * HID, CNT[t], FEATN[t]);
  }

  // ---- 3 HGT conv layers ----
  for (int L = 0; L < 3; ++L) {
    const int cb = 10 + L * 54;

    // K/Q/V projections (WMMA GEMMs, repacked B)
    for (int t = 0; t < 4; ++t) {
      float* xo = X + (size_t)OFF[t] * HID;
      int gb = (CNT[t] / 16 + 3) / 4;
      wmma_gemm_kernel<8, 0, 0><<<gb, 128, 0, stream>>>(
          xo, kwR[L][t], P(cb + 6 + t), Kb + (size_t)OFF[t] * HID, nullptr, nullptr, CNT[t]);
      wmma_gemm_kernel<8, 0, 0><<<gb, 128, 0, stream>>>(
          xo, qwR[L][t], P(cb + 34 + t), Qb + (size_t)OFF[t] * HID, nullptr, nullptr, CNT[t]);
      wmma_gemm_kernel<8, 0, 0><<<gb, 128, 0, stream>>>(
          xo, vwR[L][t], P(cb + 46 + t), Vb + (size_t)OFF[t] * HID, nullptr, nullptr, CNT[t]);
    }

    // reset segment stats + aggregation buffer
    init_kernel<<<(NTOT * HID + 255) / 256, 256, 0, stream>>>(
        AGG, AMX, ASM, NTOT * HID, NTOT * HEADS);

    // Pass A: per edge type, Ktilde = K_src @ blockdiag(a_rel); logits + seg-max
    for (int e2 = 0; e2 < 6; ++e2) {
      int s = ESRC[e2], d = EDST[e2];
      wmma_bdiag_kernel<<<(CNT[s] / 16 + 3) / 4, 128, 0, stream>>>(
          Kb + (size_t)OFF[s] * HID, arR[L][RSLOT[e2]], TIL, CNT[s]);
      edge_logits_kernel<<<egrid, 256, 0, stream>>>(
          ei[e2], TIL, Qb, OFF[d], P(cb + 28 + RSLOT[e2]),
          ALP + (size_t)e2 * EPER * HEADS, AMX);
    }

    // Pass B: exp + seg-sum
    for (int e2 = 0; e2 < 6; ++e2)
      edge_softmax_kernel<<<egrid, 256, 0, stream>>>(
          ei[e2], OFF[EDST[e2]], ALP + (size_t)e2 * EPER * HEADS, AMX, ASM);

    // Pass C: Vtilde = V_src @ blockdiag(m_rel); weighted scatter into AGG
    for (int e2 = 0; e2 < 6; ++e2) {
      int s = ESRC[e2], d = EDST[e2];
      wmma_bdiag_kernel<<<(CNT[s] / 16 + 3) / 4, 128, 0, stream>>>(
          Vb + (size_t)OFF[s] * HID, mrR[L][RSLOT[e2]], TIL, CNT[s]);
      edge_agg_kernel<<<egrid, 256, 0, stream>>>(
          ei[e2], OFF[d], ALP + (size_t)e2 * EPER * HEADS, ASM, TIL, AGG);
    }

    // o-projection with fused gelu pre-op + sigmoid-skip + leaky residual (in-place X)
    for (int t = 0; t < 4; ++t) {
      int gb = (CNT[t] / 16 + 3) / 4;
      wmma_gemm_kernel<8, 1, 1><<<gb, 128, 0, stream>>>(
          AGG + (size_t)OFF[t] * HID, owR[L][t], P(cb + 20 + t),
          nullptr, X + (size_t)OFF[t] * HID, P(cb + 42 + t), CNT[t]);
    }
  }

  // ---- output projections (128 -> 64), concatenated ----
  for (int t = 0; t < 4; ++t) {
    int gb = (CNT[t] / 16 + 3) / 4;
    wmma_gemm_kernel<4, 0, 0><<<gb, 128, 0, stream>>>(
        X + (size_t)OFF[t] * HID, outR[t], P(180 + t),
        (float*)d_out + (size_t)OFF[t] * 64, nullptr, nullptr, CNT[t]);
  }
}